// ThreeBlindMice_37752762532484
// MI455X (gfx1250) — compile-verified
//
#include <hip/hip_runtime.h>
#include <math.h>

typedef __attribute__((ext_vector_type(2))) float v2f;
typedef __attribute__((ext_vector_type(8))) float v8f;

#define NSAMP 3
#define NSYM  512
#define FDIM  20
#define NLIN  11
#define NOUT  9
#define HH    2048
#define HINN  13824   // NSAMP*NSYM*NOUT
#define NO    1536    // NSAMP*NSYM
#define EPSF  1e-9f

__device__ __forceinline__ float sigmoidf_(float x) { return 1.0f / (1.0f + expf(-x)); }

// ---------------------------------------------------------------------------
// Stage 1a: one thread per (s, b): instance-norm over F=20, 20-pt rDFT,
// bilinear(11,11,9), leaky-relu. Writes zbuf[s*27 + b*9 + k].
// Twiddles + bilinear weights live in LDS (no per-thread indexed arrays).
// ---------------------------------------------------------------------------
__global__ __launch_bounds__(256) void stage1a_kernel(
    const float* __restrict__ wax,   // [NS, S, F]
    const float* __restrict__ blw,   // [9,11,11]
    const float* __restrict__ blb,   // [9]
    float* __restrict__ zbuf)        // [S*NS*NOUT]
{
  __shared__ float sc[FDIM], ss[FDIM];
  __shared__ float sblw[NOUT * NLIN * NLIN + NOUT];
  for (int i = threadIdx.x; i < NOUT * NLIN * NLIN + NOUT; i += blockDim.x)
    sblw[i] = (i < NOUT * NLIN * NLIN) ? blw[i] : blb[i - NOUT * NLIN * NLIN];
  if (threadIdx.x < FDIM) {
    float ang = 0.314159265358979f * (float)threadIdx.x;  // 2*pi*t/20
    __sincosf(ang, &ss[threadIdx.x], &sc[threadIdx.x]);
  }
  __syncthreads();

  int tid = blockIdx.x * blockDim.x + threadIdx.x;
  if (tid >= NSYM * NSAMP) return;
  int b = tid / NSYM;          // sample
  int s = tid - b * NSYM;      // symbol

  const float* xp = wax + ((size_t)b * NSYM + s) * FDIM;
  float xv[FDIM];
  float mean = 0.f;
#pragma unroll
  for (int f = 0; f < FDIM; ++f) { xv[f] = xp[f]; mean += xv[f]; }
  mean *= (1.0f / FDIM);
  float var = 0.f;
#pragma unroll
  for (int f = 0; f < FDIM; ++f) { float d = xv[f] - mean; var += d * d; }
  var *= (1.0f / FDIM);
  float inv = rsqrtf(var + EPSF);
#pragma unroll
  for (int f = 0; f < FDIM; ++f) xv[f] = (xv[f] - mean) * inv;

  // rfft: fr[k] = sum_f xn[f] * e^{-2*pi*i*k*f/20}, k = 0..10
  float re[NLIN], im[NLIN];
#pragma unroll
  for (int k = 0; k < NLIN; ++k) {
    float rr = 0.f, ii = 0.f;
#pragma unroll
    for (int f = 0; f < FDIM; ++f) {
      int t = (k * f) % FDIM;    // constant when unrolled; else cheap int ops
      rr += xv[f] * sc[t];
      ii -= xv[f] * ss[t];
    }
    re[k] = rr; im[k] = ii;
  }

  // bilinear + leaky relu
  float* zo = zbuf + s * (NSAMP * NOUT) + b * NOUT;
#pragma unroll
  for (int k = 0; k < NOUT; ++k) {
    const float* wk = sblw + k * NLIN * NLIN;
    float acc = 0.f;
#pragma unroll
    for (int i = 0; i < NLIN; ++i) {
      float ti = 0.f;
#pragma unroll
      for (int j = 0; j < NLIN; ++j) ti += wk[i * NLIN + j] * im[j];
      acc += re[i] * ti;
    }
    acc += sblw[NOUT * NLIN * NLIN + k];
    zo[k] = (acc > 0.f) ? acc : 0.01f * acc;
  }
}

// ---------------------------------------------------------------------------
// Stage 1b: softmax over the 3 samples, per (s, k). 4608 threads.
// ---------------------------------------------------------------------------
__global__ void stage1b_kernel(const float* __restrict__ zbuf,
                               float* __restrict__ v)
{
  int idx = blockIdx.x * blockDim.x + threadIdx.x;   // s*9 + k
  if (idx >= NSYM * NOUT) return;
  int s = idx / NOUT;
  int k = idx - s * NOUT;
  int base = s * (NSAMP * NOUT) + k;
  float a0 = zbuf[base];
  float a1 = zbuf[base + NOUT];
  float a2 = zbuf[base + 2 * NOUT];
  float mx = fmaxf(a0, fmaxf(a1, a2));
  float e0 = expf(a0 - mx), e1 = expf(a1 - mx), e2 = expf(a2 - mx);
  float rs = 1.0f / (e0 + e1 + e2);
  v[base]            = e0 * rs;
  v[base + NOUT]     = e1 * rs;
  v[base + 2 * NOUT] = e2 * rs;
}

// ---------------------------------------------------------------------------
// Bandwidth-bound GEMV via V_WMMA_F32_16X16X4_F32 (full f32 precision).
// One workgroup (8 waves) per 16-row tile; waves split K; x staged in LDS.
// A layout (ISA 16x4 f32): lane m=lane&15, contiguous float2 at k+2*(lane>>4).
// B: x replicated across 16 columns -> same contiguous float2 pattern.
// Manual 4x unroll, two alternating accumulators (4 loads in flight, split
// D->C dependency chain). D column 0: lane 0 has M=0..7, lane 16 M=8..15.
// remapH!=0 skips dead f-gate rows: logical [H,2H) -> actual [2H,3H) etc.
// ---------------------------------------------------------------------------
__global__ __launch_bounds__(256) void gemv_wmma_kernel(
    const float* __restrict__ W,      // [rows, K] row-major
    const float* __restrict__ bias1,
    const float* __restrict__ bias2,  // may be nullptr
    const float* __restrict__ x,      // [K]
    float* __restrict__ out,          // [gridDim.x*16]
    int K, int remapH)
{
  extern __shared__ float smem[];
  float* xs  = smem;        // K floats
  float* red = smem + K;    // 128 floats

  int tid = threadIdx.x;
  const float4* x4  = (const float4*)x;
  float4*       xs4 = (float4*)xs;
  for (int i = tid; i < (K >> 2); i += blockDim.x) xs4[i] = x4[i];
  __syncthreads();

  int wave  = tid >> 5;
  int lane  = tid & 31;
  int mlane = lane & 15;
  int khalf = lane >> 4;

  int tbase = blockIdx.x * 16;
  int abase = tbase + ((remapH != 0 && tbase >= remapH) ? remapH : 0);

  const float* wrow = W + (size_t)(abase + mlane) * (size_t)K + 2 * khalf;
  const float* xrow = xs + 2 * khalf;
  int kw = K >> 3;           // per-wave K slice; divisible by 16 for all uses
  int k0 = wave * kw;
  int k1 = k0 + kw;

  v8f accA = {}, accB = {};
  for (int k = k0; k < k1; k += 16) {
    float2 a0 = *(const float2*)(wrow + k);
    float2 a1 = *(const float2*)(wrow + k + 4);
    float2 a2 = *(const float2*)(wrow + k + 8);
    float2 a3 = *(const float2*)(wrow + k + 12);
    float2 b0 = *(const float2*)(xrow + k);
    float2 b1 = *(const float2*)(xrow + k + 4);
    float2 b2 = *(const float2*)(xrow + k + 8);
    float2 b3 = *(const float2*)(xrow + k + 12);
    v2f av0; av0.x = a0.x; av0.y = a0.y;
    v2f av1; av1.x = a1.x; av1.y = a1.y;
    v2f av2; av2.x = a2.x; av2.y = a2.y;
    v2f av3; av3.x = a3.x; av3.y = a3.y;
    v2f bv0; bv0.x = b0.x; bv0.y = b0.y;
    v2f bv1; bv1.x = b1.x; bv1.y = b1.y;
    v2f bv2; bv2.x = b2.x; bv2.y = b2.y;
    v2f bv3; bv3.x = b3.x; bv3.y = b3.y;
    accA = __builtin_amdgcn_wmma_f32_16x16x4_f32(false, av0, false, bv0, (short)0, accA, false, false);
    accB = __builtin_amdgcn_wmma_f32_16x16x4_f32(false, av1, false, bv1, (short)0, accB, false, false);
    accA = __builtin_amdgcn_wmma_f32_16x16x4_f32(false, av2, false, bv2, (short)0, accA, false, false);
    accB = __builtin_amdgcn_wmma_f32_16x16x4_f32(false, av3, false, bv3, (short)0, accB, false, false);
  }
  v8f acc = accA + accB;

  if (lane == 0) {
#pragma unroll
    for (int g = 0; g < 8; ++g) red[wave * 16 + g] = acc[g];
  }
  if (lane == 16) {
#pragma unroll
    for (int g = 0; g < 8; ++g) red[wave * 16 + 8 + g] = acc[g];
  }
  __syncthreads();

  if (tid < 16) {
    float ssum = 0.f;
#pragma unroll
    for (int w = 0; w < 8; ++w) ssum += red[w * 16 + tid];
    int a = abase + tid;
    ssum += bias1[a];
    if (bias2) ssum += bias2[a];
    out[tbase + tid] = ssum;
  }
}

// ---------------------------------------------------------------------------
// LSTM activation with h0=c0=0: h = sigmoid(o) * tanh(sigmoid(i) * tanh(g)).
// gates layout: [0,H)=i, [H,2H)=g, [2H,3H)=o (f-gate rows never computed).
// ---------------------------------------------------------------------------
__global__ void lstm_act_kernel(const float* __restrict__ g, float* __restrict__ h)
{
  int j = blockIdx.x * blockDim.x + threadIdx.x;
  if (j >= HH) return;
  float ig = g[j], gg = g[HH + j], og = g[2 * HH + j];
  float c = sigmoidf_(ig) * tanhf(gg);
  h[j] = sigmoidf_(og) * tanhf(c);
}

// ---------------------------------------------------------------------------
// Finalize: per row of [NS, S]: instance-norm over S, leaky-relu, softmax.
// One block (256 threads, 2 elems/thread) per row.
// ---------------------------------------------------------------------------
__global__ __launch_bounds__(256) void finalize_kernel(
    const float* __restrict__ o, float* __restrict__ out)
{
  __shared__ float rbuf[256];
  int t = threadIdx.x;
  int base = blockIdx.x * NSYM;
  float x0 = o[base + t], x1 = o[base + t + 256];

  rbuf[t] = x0 + x1; __syncthreads();
  for (int off = 128; off > 0; off >>= 1) { if (t < off) rbuf[t] += rbuf[t + off]; __syncthreads(); }
  float mean = rbuf[0] * (1.0f / NSYM); __syncthreads();

  float d0 = x0 - mean, d1 = x1 - mean;
  rbuf[t] = d0 * d0 + d1 * d1; __syncthreads();
  for (int off = 128; off > 0; off >>= 1) { if (t < off) rbuf[t] += rbuf[t + off]; __syncthreads(); }
  float var = rbuf[0] * (1.0f / NSYM); __syncthreads();

  float inv = rsqrtf(var + EPSF);
  float y0 = d0 * inv, y1 = d1 * inv;
  y0 = (y0 > 0.f) ? y0 : 0.01f * y0;
  y1 = (y1 > 0.f) ? y1 : 0.01f * y1;

  rbuf[t] = fmaxf(y0, y1); __syncthreads();
  for (int off = 128; off > 0; off >>= 1) { if (t < off) rbuf[t] = fmaxf(rbuf[t], rbuf[t + off]); __syncthreads(); }
  float mx = rbuf[0]; __syncthreads();

  float e0 = expf(y0 - mx), e1 = expf(y1 - mx);
  rbuf[t] = e0 + e1; __syncthreads();
  for (int off = 128; off > 0; off >>= 1) { if (t < off) rbuf[t] += rbuf[t + off]; __syncthreads(); }
  float rs = 1.0f / rbuf[0];

  out[base + t]       = e0 * rs;
  out[base + t + 256] = e1 * rs;
}

// ---------------------------------------------------------------------------
extern "C" void kernel_launch(void* const* d_in, const int* in_sizes, int n_in,
                              void* d_out, int out_size, void* d_ws, size_t ws_size,
                              hipStream_t stream)
{
  const float* wax   = (const float*)d_in[0];
  const float* blw   = (const float*)d_in[1];
  const float* blb   = (const float*)d_in[2];
  const float* w_ih0 = (const float*)d_in[3];
  /* d_in[4] = w_hh0: dead (h0 == 0) */
  const float* b_ih0 = (const float*)d_in[5];
  const float* b_hh0 = (const float*)d_in[6];
  const float* w_ih1 = (const float*)d_in[7];
  const float* b_ih1 = (const float*)d_in[9];
  const float* b_hh1 = (const float*)d_in[10];
  const float* w_ih2 = (const float*)d_in[11];
  const float* b_ih2 = (const float*)d_in[13];
  const float* b_hh2 = (const float*)d_in[14];
  const float* lin_w = (const float*)d_in[15];
  const float* lin_b = (const float*)d_in[16];
  float* out = (float*)d_out;

  float* ws    = (float*)d_ws;
  float* v     = ws;                      // 13824
  float* zbuf  = v + HINN;                // 13824
  float* gates = zbuf + HINN;             // 6144 (i,g,o pre-activations)
  float* hA    = gates + 3 * HH;          // 2048
  float* hB    = hA + HH;                 // 2048
  float* opre  = hB + HH;                 // 1536  (total ~154 KB of d_ws)
  (void)in_sizes; (void)n_in; (void)out_size; (void)ws_size;

  stage1a_kernel<<<(NSYM * NSAMP + 255) / 256, 256, 0, stream>>>(wax, blw, blb, zbuf);
  stage1b_kernel<<<(NSYM * NOUT + 255) / 256, 256, 0, stream>>>(zbuf, v);

  size_t smBig   = (size_t)HINN * 4 + 128 * 4;   // 55.8 KB (fits 320KB/WGP LDS)
  size_t smSmall = (size_t)HH   * 4 + 128 * 4;   // 8.7 KB

  // cell 0: 6144 live gate rows (i,g,o), K = 13824
  gemv_wmma_kernel<<<(3 * HH) / 16, 256, smBig, stream>>>(w_ih0, b_ih0, b_hh0, v, gates, HINN, HH);
  lstm_act_kernel<<<HH / 256, 256, 0, stream>>>(gates, hA);
  // cell 1
  gemv_wmma_kernel<<<(3 * HH) / 16, 256, smSmall, stream>>>(w_ih1, b_ih1, b_hh1, hA, gates, HH, HH);
  lstm_act_kernel<<<HH / 256, 256, 0, stream>>>(gates, hB);
  // cell 2
  gemv_wmma_kernel<<<(3 * HH) / 16, 256, smSmall, stream>>>(w_ih2, b_ih2, b_hh2, hB, gates, HH, HH);
  lstm_act_kernel<<<HH / 256, 256, 0, stream>>>(gates, hA);
  // final linear: 1536 rows, no remap, single bias
  gemv_wmma_kernel<<<NO / 16, 256, smSmall, stream>>>(lin_w, lin_b, nullptr, hA, opre, HH, 0);

  finalize_kernel<<<NSAMP, 256, 0, stream>>>(opre, out);
}